// Jitter_88716844466943
// MI455X (gfx1250) — compile-verified
//
#include <hip/hip_runtime.h>
#include <stdint.h>

#define B_DIM  32
#define I_DIM  512
#define T_DIM  4096
#define NSTEP  (T_DIM - 2)      /* 4094 Markov steps */
#define TILE   1024
#define HALO   4                /* keeps B128 LDS destinations 16B-aligned */

#if defined(__HIP_DEVICE_COMPILE__) && defined(__gfx1250__) &&              \
    __has_builtin(__builtin_amdgcn_global_load_async_to_lds_b128) &&        \
    __has_builtin(__builtin_amdgcn_global_load_async_to_lds_b32)  &&        \
    __has_builtin(__builtin_amdgcn_s_wait_asynccnt)
#define HAVE_ASYNC 1
#else
#define HAVE_ASYNC 0
#endif

// ---------------- threefry2x32 (JAX PRNG core) ----------------
__device__ __forceinline__ uint32_t rotl32(uint32_t v, int n) {
  return (v << n) | (v >> (32 - n));
}

__device__ __forceinline__ void threefry2x32(uint32_t k0, uint32_t k1,
                                             uint32_t x0, uint32_t x1,
                                             uint32_t& y0, uint32_t& y1) {
  const uint32_t ks0 = k0, ks1 = k1, ks2 = k0 ^ k1 ^ 0x1BD11BDAu;
  x0 += ks0; x1 += ks1;
#define TF_R(r) { x0 += x1; x1 = rotl32(x1, (r)); x1 ^= x0; }
  TF_R(13) TF_R(15) TF_R(26) TF_R(6)
  x0 += ks1; x1 += ks2 + 1u;
  TF_R(17) TF_R(29) TF_R(16) TF_R(24)
  x0 += ks2; x1 += ks0 + 2u;
  TF_R(13) TF_R(15) TF_R(26) TF_R(6)
  x0 += ks0; x1 += ks1 + 3u;
  TF_R(17) TF_R(29) TF_R(16) TF_R(24)
  x0 += ks1; x1 += ks2 + 4u;
  TF_R(13) TF_R(15) TF_R(26) TF_R(6)
  x0 += ks2; x1 += ks0 + 5u;
#undef TF_R
  y0 = x0; y1 = x1;
}

// uniform-in-[tiny,1) -> gumbel, exactly as jax.random._uniform/gumbel
__device__ __forceinline__ float bits_to_gumbel(uint32_t bits) {
  const float TINY = 1.17549435e-38f;  // FLT_MIN
  float f = __uint_as_float((bits >> 9) | 0x3f800000u) - 1.0f;
  float u = fmaxf(TINY, f * (1.0f - TINY) + TINY);
  return -logf(-logf(u));
}

// ---------------- K1: per-step keys (partitionable split) ----------------
__global__ void jitter_keys_kernel(uint32_t* __restrict__ keyout) {
  const int t = blockIdx.x * blockDim.x + threadIdx.x;
  if (t >= NSTEP) return;
  uint32_t y0, y1;
  threefry2x32(0u, 42u, 0u, (uint32_t)t, y0, y1);   // key(42) = (0,42)
  keyout[2 * t]     = y0;
  keyout[2 * t + 1] = y1;
}

// ---------------- K2: per-(t,b) gumbels + 9-state argmax decisions -------
__global__ void jitter_decide_kernel(const uint32_t* __restrict__ keyout,
                                     uint32_t* __restrict__ cw) {
  const int gid = blockIdx.x * blockDim.x + threadIdx.x;
  if (gid >= NSTEP * B_DIM) return;
  const int t = gid >> 5;
  const int b = gid & 31;
  const uint32_t k0 = keyout[2 * t], k1 = keyout[2 * t + 1];

  float gg[3];
#pragma unroll
  for (int c = 0; c < 3; ++c) {
    uint32_t y0, y1;
    threefry2x32(k0, k1, 0u, (uint32_t)(3 * b + c), y0, y1);
    gg[c] = bits_to_gumbel(y0 ^ y1);   // partitionable 32-bit draw
  }

  const float LP  = -2.30258509f;      // log 0.1
  const float LS  = -0.223143551f;     // log 0.8
  const float L89 = -0.117783035f;     // log(8/9)
  const float L19 = -2.19722458f;      // log(1/9)

  uint32_t w = 0;
#pragma unroll
  for (int st = 0; st < 9; ++st) {     // st = p2*3 + p1
    const int p2 = st / 3, p1 = st % 3;
    float l0, l1, l2;
    if (p2 == 1 && p1 == 2) { l0 = -__builtin_inff(); l1 = L89; l2 = L19; }
    else                    { l0 = LP;                l1 = LS;  l2 = LP;  }
    float s0 = l0 + gg[0], s1 = l1 + gg[1], s2 = l2 + gg[2];
    int s = 0; float best = s0;
    if (s1 > best) { best = s1; s = 1; }   // strict > : first-max like argmax
    if (s2 > best) { s = 2; }
    w |= (uint32_t)s << (2 * st);
  }
  cw[gid] = w;
}

// ---------------- K3: 32-lane integer recurrence -> offsets d[b][t] ------
__global__ void jitter_chain_kernel(const uint32_t* __restrict__ cw,
                                    int* __restrict__ dmask) {
  const int b = threadIdx.x;           // 32 lanes, one wave
  if (b >= B_DIM) return;
  int st = 4;                          // (p2,p1) = (1,1)
  dmask[b * T_DIM + 0] = 0;            // endpoints fixed: m=1 -> d=0
  for (int tb = 0; tb < NSTEP; tb += 16) {
    uint32_t wreg[16];
#pragma unroll
    for (int j = 0; j < 16; ++j) {     // batch loads -> pipelined latency
      const int t = tb + j;
      wreg[j] = (t < NSTEP) ? cw[t * 32 + b] : 0u;
    }
#pragma unroll
    for (int j = 0; j < 16; ++j) {
      const int t = tb + j;
      if (t < NSTEP) {
        const int s = (int)((wreg[j] >> (2 * st)) & 3u);
        dmask[b * T_DIM + t + 1] = s - 1;
        st = (st % 3) * 3 + s;         // (p2,p1) <- (p1,s)
      }
    }
  }
  dmask[b * T_DIM + (T_DIM - 1)] = 0;
}

// ---------------- K4: memory-bound jitter gather (async LDS staging) -----
#if HAVE_ASYNC
typedef int v4i __attribute__((vector_size(16)));
#define AS1 __attribute__((address_space(1)))
#define AS3 __attribute__((address_space(3)))
#endif

__global__ __launch_bounds__(256) void jitter_gather_kernel(
    const float* __restrict__ x, const int* __restrict__ dmask,
    float* __restrict__ out) {
  __shared__ __align__(16) float lx[TILE + 2 * HALO];
  const int tid = threadIdx.x;
  const unsigned blk = blockIdx.x;
  const int seg = (int)(blk & (T_DIM / TILE - 1));
  const int row = (int)(blk >> 2);          // b*I_DIM + i
  const int b   = row >> 9;                 // / I_DIM
  const long long roff = (long long)row * T_DIM;
  const float* __restrict__ xrow = x + roff;
  const int t0 = seg * TILE;
  const int lo = (t0 > 0) ? (t0 - 1) : 0;                          // clamped halo
  const int hi = (t0 + TILE < T_DIM) ? (t0 + TILE) : (T_DIM - 1);  // slots never
                                                                   // read when clamped
#if HAVE_ASYNC
  __builtin_amdgcn_global_load_async_to_lds_b128(
      (AS1 v4i*)(xrow + t0 + 4 * tid),
      (AS3 v4i*)&lx[HALO + 4 * tid], 0, 0);
  if (tid == 0)
    __builtin_amdgcn_global_load_async_to_lds_b32(
        (AS1 int*)(xrow + lo), (AS3 int*)&lx[HALO - 1], 0, 0);
  else if (tid == 1)
    __builtin_amdgcn_global_load_async_to_lds_b32(
        (AS1 int*)(xrow + hi), (AS3 int*)&lx[HALO + TILE], 0, 0);
  __builtin_amdgcn_s_wait_asynccnt(0);
  __syncthreads();
#else
  *(float4*)&lx[HALO + 4 * tid] = *(const float4*)(xrow + t0 + 4 * tid);
  if (tid == 0)      lx[HALO - 1]    = xrow[lo];
  else if (tid == 1) lx[HALO + TILE] = xrow[hi];
  __syncthreads();
#endif

  const int4 dv = *(const int4*)(dmask + b * T_DIM + t0 + 4 * tid);
  const int base = HALO + 4 * tid;
  float4 v;
  v.x = lx[base + 0 + dv.x];
  v.y = lx[base + 1 + dv.y];
  v.z = lx[base + 2 + dv.z];
  v.w = lx[base + 3 + dv.w];
  *(float4*)(out + roff + t0 + 4 * tid) = v;
}

// ---------------- launch ----------------
extern "C" void kernel_launch(void* const* d_in, const int* in_sizes, int n_in,
                              void* d_out, int out_size, void* d_ws, size_t ws_size,
                              hipStream_t stream) {
  (void)in_sizes; (void)n_in; (void)out_size; (void)ws_size;
  const float* x = (const float*)d_in[0];
  float* out = (float*)d_out;

  // ws layout (u32 words): keys[2*NSTEP] | cw[NSTEP*32] | dmask[32*4096]  (~1.0 MB)
  uint32_t* keyout = (uint32_t*)d_ws;
  uint32_t* cw     = keyout + 8192;
  int*      dmask  = (int*)(cw + (size_t)NSTEP * B_DIM);

  jitter_keys_kernel  <<<(NSTEP + 255) / 256, 256, 0, stream>>>(keyout);
  jitter_decide_kernel<<<(NSTEP * B_DIM + 255) / 256, 256, 0, stream>>>(keyout, cw);
  jitter_chain_kernel <<<1, 32, 0, stream>>>(cw, dmask);
  jitter_gather_kernel<<<B_DIM * I_DIM * (T_DIM / TILE), 256, 0, stream>>>(x, dmask, out);
}